// CapsuleNet_45294725103875
// MI455X (gfx1250) — compile-verified
//
#include <hip/hip_runtime.h>

// CapsuleNet forward for MI455X (gfx1250, wave32).
// Tile = 16 batch elements per workgroup (256 threads / 8 waves).
//  - hidden-feature staging to LDS via TDM tensor_load_to_lds (async tensor path,
//    LDS row padding 256+32 dwords -> stride-288 rows), fenced by s_wait_tensorcnt.
//  - 1x1 conv and x_hat GEMMs on V_WMMA_F32_16X16X4_F32.
//  - x_hat cached in LDS as fp16 (198 KB of the 320 KB WGP LDS); routing from LDS.

#define L_TOK    10
#define D_EMB    8
#define IN_CAPS  36
#define OUT_CAPS 11
#define OUT_DIM  16
#define TB       16
#define NTHREADS 256

typedef __attribute__((ext_vector_type(2))) float        v2f;
typedef __attribute__((ext_vector_type(8))) float        v8f;
typedef __attribute__((ext_vector_type(8))) _Float16     v8h;
typedef __attribute__((ext_vector_type(4))) unsigned int u32x4;
typedef __attribute__((ext_vector_type(8))) unsigned int u32x8;

__global__ __launch_bounds__(NTHREADS)
void capsnet_fwd(const float* __restrict__ hidden,
                 const int*   __restrict__ e1_tok, const int* __restrict__ e2_tok,
                 const int*   __restrict__ e1_typ, const int* __restrict__ e2_typ,
                 const int*   __restrict__ e1_len, const int* __restrict__ e2_len,
                 const float* __restrict__ type_emb, const float* __restrict__ ent_emb,
                 const float* __restrict__ att_w,
                 const float* __restrict__ conv_w, const float* __restrict__ conv_b,
                 const float* __restrict__ caps_w,
                 float* __restrict__ out)
{
    __shared__ float    s_xin[TB * 288];                              // concat feats; reused as routing c
    __shared__ float    s_x  [TB * 288];                              // primary caps [b][cap*8+j]
    __shared__ _Float16 s_xh [OUT_CAPS * IN_CAPS * TB * OUT_DIM];     // x_hat fp16, [o][i][b][D]
    __shared__ float    s_bl [TB * OUT_CAPS * IN_CAPS];               // routing logits
    float* s_c = s_xin;                                               // overlay: c coefficients

    const int tid  = threadIdx.x;
    const int lane = tid & 31;
    const int wave = tid >> 5;
    const int b0   = blockIdx.x * TB;

    // Prefetch caps_w (shared across all blocks, L2-resident) : global_prefetch_b8.
    for (int p = tid; p < OUT_CAPS * IN_CAPS; p += NTHREADS)
        __builtin_prefetch(caps_w + p * 128, 0, 1);

    // ---------------- Phase 0a: TDM async copy hidden[b0:b0+16, 0:256] -> s_xin rows.
    // D# (CDNA5 ISA): 2D tensor, 4B elems, tile 256x16,
    // pad_enable: after every 256 dwords stored, skip 32 dwords -> LDS row stride 288.
    if (wave == 0) {
        const unsigned           lds_off = (unsigned)(uintptr_t)(&s_xin[0]);
        const unsigned long long ga      = (unsigned long long)(uintptr_t)(hidden + (size_t)b0 * 256);
        u32x4 g0;
        g0[0] = 1u;                                                   // count=1, user descriptor
        g0[1] = lds_off;                                              // lds_addr
        g0[2] = (unsigned)ga;                                         // global_addr[31:0]
        g0[3] = (unsigned)((ga >> 32) & 0x1FFFFFFu) | (2u << 30);     // global_addr[56:32] | type=2
        u32x8 g1;
        g1[0] = (2u << 16) | (1u << 20) | (7u << 22) | (31u << 25);   // data_size=4B, pad_en, ivl=256dw, amt=32dw
        g1[1] = (256u & 0xFFFFu) << 16;                               // tensor_dim0 = 256
        g1[2] = (16u << 16);                                          // tensor_dim1 = 16
        g1[3] = (256u << 16);                                         // tile_dim0 = 256
        g1[4] = 16u;                                                  // tile_dim1 = 16, tile_dim2 = 0
        g1[5] = 256u;                                                 // tensor_dim0_stride = 256
        g1[6] = 0u;
        g1[7] = 0u;
        asm volatile("tensor_load_to_lds %0, %1" :: "s"(g0), "s"(g1) : "memory");
    }

    // ---------------- Phase 0b: embedding gathers + attention pooling -> s_xin[b][256..287]
    {
        const int bl_ = tid >> 4;      // local batch 0..15
        const int sub = tid & 15;      // 16 helper threads per batch element
        const int bg  = b0 + bl_;

        if (sub == 0) {
            const float* te = type_emb + e1_typ[bg] * D_EMB;
            #pragma unroll
            for (int j = 0; j < 8; ++j) s_xin[bl_ * 288 + 256 + j] = te[j];
        } else if (sub == 1) {
            const float* te = type_emb + e2_typ[bg] * D_EMB;
            #pragma unroll
            for (int j = 0; j < 8; ++j) s_xin[bl_ * 288 + 264 + j] = te[j];
        } else if (sub == 2 || sub == 3) {
            // dot-product attention pooling over entity mention tokens
            const int* tok = (sub == 2 ? e1_tok : e2_tok) + bg * L_TOK;
            const int  len = (sub == 2 ? e1_len : e2_len)[bg];
            float aw[8];
            #pragma unroll
            for (int d = 0; d < 8; ++d) aw[d] = att_w[d];
            float sc[L_TOK];
            float m = -1e30f;
            #pragma unroll
            for (int l = 0; l < L_TOK; ++l) {
                const float* em = ent_emb + (size_t)tok[l] * D_EMB;
                float s = 0.f;
                #pragma unroll
                for (int d = 0; d < 8; ++d) s += em[d] * aw[d];
                s = (l < len) ? s : -1e9f;
                sc[l] = s;
                m = fmaxf(m, s);
            }
            float ssum = 0.f;
            #pragma unroll
            for (int l = 0; l < L_TOK; ++l) { sc[l] = __expf(sc[l] - m); ssum += sc[l]; }
            const float inv = 1.f / ssum;
            float acc[8] = {0.f, 0.f, 0.f, 0.f, 0.f, 0.f, 0.f, 0.f};
            #pragma unroll
            for (int l = 0; l < L_TOK; ++l) {
                const float* em = ent_emb + (size_t)tok[l] * D_EMB;
                const float a = sc[l] * inv;
                #pragma unroll
                for (int d = 0; d < 8; ++d) acc[d] += a * em[d];
            }
            const int base = bl_ * 288 + (sub == 2 ? 272 : 280);
            #pragma unroll
            for (int d = 0; d < 8; ++d) s_xin[base + d] = acc[d];
        }
    }
    __builtin_amdgcn_s_wait_tensorcnt(0);   // TDM tile landed (wave0; no-op elsewhere)
    __syncthreads();

    // ---------------- Phase 1a: 1x1 conv as WMMA: [288 rows(b,pos) x 16ch] @ [16ch x 16o]
    {
        const bool hi   = lane >= 16;
        const int  nidx = lane & 15;          // A row-in-tile / B,C column (out channel)
        for (int t = wave; t < 18; t += NTHREADS / 32) {
            const int row = t * 16 + nidx;    // A-operand row for this lane
            const int ab  = row / 18, apos = row % 18;
            v8f c8 = {};
            #pragma unroll
            for (int s = 0; s < 4; ++s) {
                const int c0 = 4 * s + (hi ? 2 : 0);
                v2f a  = { s_xin[ab * 288 + c0 * 18 + apos],
                           s_xin[ab * 288 + (c0 + 1) * 18 + apos] };
                v2f bm = { conv_w[nidx * 16 + c0], conv_w[nidx * 16 + c0 + 1] };
                c8 = __builtin_amdgcn_wmma_f32_16x16x4_f32(false, a, false, bm, (short)0, c8, false, false);
            }
            const float bias = conv_b[nidx];
            #pragma unroll
            for (int r = 0; r < 8; ++r) {
                const int rr = t * 16 + r + (hi ? 8 : 0);
                const int b = rr / 18, pos = rr % 18;
                s_x[b * 288 + nidx * 18 + pos] = c8[r] + bias;
            }
        }
    }
    __syncthreads();

    // squash each 8-dim primary capsule in place
    for (int k = tid; k < TB * IN_CAPS; k += NTHREADS) {
        float* v = &s_x[k * 8];
        float tmp[8];
        float n2 = 0.f;
        #pragma unroll
        for (int j = 0; j < 8; ++j) { tmp[j] = v[j]; n2 += tmp[j] * tmp[j]; }
        const float nn = sqrtf(n2);
        const float sc = n2 / (1.f + n2) / (nn + 1e-8f);
        #pragma unroll
        for (int j = 0; j < 8; ++j) v[j] = sc * tmp[j];
    }
    __syncthreads();

    // ---------------- Phase 1b: x_hat[o][i] = x_i[16bx8] @ W_oi^T[8x16] via WMMA f32 K=4x2
    {
        const bool hi   = lane >= 16;
        const int  nidx = lane & 15;          // A row (batch) / B,C column (D)
        const int  dsel = hi ? 2 : 0;         // K-slot pair per lane half
        for (int t = wave; t < OUT_CAPS * IN_CAPS; t += NTHREADS / 32) {
            const int i = t % IN_CAPS;
            // A: rows = batch, K = d  (lanes 0-15: K{0,1}, lanes 16-31: K{2,3}; step2: +4)
            const float* xp = &s_x[nidx * 288 + i * 8 + dsel];
            // B: [k=d][n=D] = caps_w[o][i][D][d]
            const float* wp = caps_w + ((size_t)t * 16 + nidx) * 8 + dsel;
            v2f a0  = { xp[0], xp[1] };
            v2f a1  = { xp[4], xp[5] };
            v2f bm0 = { wp[0], wp[1] };
            v2f bm1 = { wp[4], wp[5] };
            v8f c = {};
            c = __builtin_amdgcn_wmma_f32_16x16x4_f32(false, a0, false, bm0, (short)0, c, false, false);
            c = __builtin_amdgcn_wmma_f32_16x16x4_f32(false, a1, false, bm1, (short)0, c, false, false);
            // C tile: vgpr r -> batch row r (+8 for hi lanes), col = nidx (D)
            _Float16* dst = &s_xh[(size_t)t * TB * OUT_DIM];
            #pragma unroll
            for (int r = 0; r < 8; ++r) {
                const int brow = r + (hi ? 8 : 0);
                dst[brow * OUT_DIM + nidx] = (_Float16)c[r];
            }
        }
    }
    __syncthreads();

    // ---------------- Phase 2: dynamic routing (3 iters) from LDS
    for (int iter = 0; iter < 3; ++iter) {
        if (iter > 0) {
            // c = softmax over out-caps of logits, per (b, in-cap)
            for (int idx = tid; idx < TB * IN_CAPS; idx += NTHREADS) {
                const int b = idx / IN_CAPS, i = idx % IN_CAPS;
                float vals[OUT_CAPS];
                float m = -1e30f;
                #pragma unroll
                for (int o = 0; o < OUT_CAPS; ++o) {
                    vals[o] = s_bl[(b * OUT_CAPS + o) * IN_CAPS + i];
                    m = fmaxf(m, vals[o]);
                }
                float ssum = 0.f;
                #pragma unroll
                for (int o = 0; o < OUT_CAPS; ++o) { vals[o] = __expf(vals[o] - m); ssum += vals[o]; }
                const float inv = 1.f / ssum;
                #pragma unroll
                for (int o = 0; o < OUT_CAPS; ++o)
                    s_c[(b * OUT_CAPS + o) * IN_CAPS + i] = vals[o] * inv;
            }
            __syncthreads();
        }
        for (int q = tid; q < TB * OUT_CAPS; q += NTHREADS) {
            const int b = q / OUT_CAPS, o = q % OUT_CAPS;
            float acc[16];
            #pragma unroll
            for (int D = 0; D < 16; ++D) acc[D] = 0.f;
            for (int i = 0; i < IN_CAPS; ++i) {
                const float cw = (iter == 0) ? (1.0f / 11.0f)
                                             : s_c[(b * OUT_CAPS + o) * IN_CAPS + i];
                const v8h* xp = (const v8h*)&s_xh[((size_t)(o * IN_CAPS + i) * TB + b) * OUT_DIM];
                const v8h h0 = xp[0], h1 = xp[1];
                #pragma unroll
                for (int D = 0; D < 8; ++D) acc[D]     += cw * (float)h0[D];
                #pragma unroll
                for (int D = 0; D < 8; ++D) acc[8 + D] += cw * (float)h1[D];
            }
            float n2 = 0.f;
            #pragma unroll
            for (int D = 0; D < 16; ++D) n2 += acc[D] * acc[D];
            const float nn = sqrtf(n2);
            const float sc = n2 / (1.f + n2) / (nn + 1e-8f);
            if (iter == 2) {
                out[(size_t)(b0 + b) * OUT_CAPS + o] = sc * nn;   // ||squash(s)||
            } else {
                // logits update: b += <v, x_hat_i>, v = sc * s
                for (int i = 0; i < IN_CAPS; ++i) {
                    const v8h* xp = (const v8h*)&s_xh[((size_t)(o * IN_CAPS + i) * TB + b) * OUT_DIM];
                    const v8h h0 = xp[0], h1 = xp[1];
                    float d = 0.f;
                    #pragma unroll
                    for (int D = 0; D < 8; ++D) d += acc[D]     * (float)h0[D];
                    #pragma unroll
                    for (int D = 0; D < 8; ++D) d += acc[8 + D] * (float)h1[D];
                    d *= sc;
                    float* bp = &s_bl[(b * OUT_CAPS + o) * IN_CAPS + i];
                    *bp = (iter == 0) ? d : (*bp + d);
                }
            }
        }
        __syncthreads();
    }
}

extern "C" void kernel_launch(void* const* d_in, const int* in_sizes, int n_in,
                              void* d_out, int out_size, void* d_ws, size_t ws_size,
                              hipStream_t stream) {
    const float* hidden   = (const float*)d_in[0];
    const int*   e1_tok   = (const int*)  d_in[1];
    const int*   e2_tok   = (const int*)  d_in[2];
    const int*   e1_typ   = (const int*)  d_in[3];
    const int*   e2_typ   = (const int*)  d_in[4];
    const int*   e1_len   = (const int*)  d_in[5];
    const int*   e2_len   = (const int*)  d_in[6];
    const float* type_emb = (const float*)d_in[7];
    const float* ent_emb  = (const float*)d_in[8];
    const float* att_w    = (const float*)d_in[9];
    const float* conv_w   = (const float*)d_in[10];
    const float* conv_b   = (const float*)d_in[11];
    const float* caps_w   = (const float*)d_in[12];
    float*       out      = (float*)d_out;

    const int B = in_sizes[0] / 256;           // 8192
    const int nblocks = B / TB;                // 512
    capsnet_fwd<<<nblocks, NTHREADS, 0, stream>>>(
        hidden, e1_tok, e2_tok, e1_typ, e2_typ, e1_len, e2_len,
        type_emb, ent_emb, att_w, conv_w, conv_b, caps_w, out);
}